// DenseGrid_88278757802386
// MI455X (gfx1250) — compile-verified
//
#include <hip/hip_runtime.h>
#include <hip/hip_bf16.h>
#include <stdint.h>

// -------------------------------------------------------------------------
// Multi-LOD dense-grid trilinear interpolation for MI455X (gfx1250).
// Gather-bound: exploit 192MB L2 (all codebooks resident), NT hints for the
// streamed points/output, and TDM (tensor_load_to_lds) to stage the two
// small LODs (16^3 + 32^3 = 144 KB) into LDS per workgroup.
// -------------------------------------------------------------------------

typedef unsigned int u32x4 __attribute__((ext_vector_type(4)));
typedef int          i32x4 __attribute__((ext_vector_type(4)));
typedef int          i32x8 __attribute__((ext_vector_type(8)));

#define BLOCK_SIZE 512

#if defined(__has_builtin)
#  if __has_builtin(__builtin_amdgcn_tensor_load_to_lds)
#    define HAVE_TDM 1
#  endif
#endif
#ifndef HAVE_TDM
#  define HAVE_TDM 0
#endif

// Build a D# for a contiguous 2D f32 tile (dim0 x dim1 elements, stride=dim0)
// and issue TENSOR_LOAD_TO_LDS.  Descriptor layout per CDNA5 ISA §8:
//   group0: [1:0]=count=1, [63:32]=lds_addr, [120:64]=global_addr, [127:126]=2
//   group1: data_size=2(4B) @ [17:16]; tensor_dim0 @ [79:48]; tensor_dim1 @
//           [111:80]; tile_dim0 @ [127:112]; tile_dim1 @ [143:128];
//           tensor_dim0_stride @ [207:160]
__device__ __forceinline__ void tdm_load_2d_f32(const float* gptr,
                                                unsigned lds_byte,
                                                unsigned dim0, unsigned dim1) {
#if HAVE_TDM
  unsigned long long ga = (unsigned long long)(uintptr_t)gptr;
  u32x4 g0;
  g0.x = 1u;                                              // count=1, user mode
  g0.y = lds_byte;                                        // lds_addr (bytes)
  g0.z = (unsigned)(ga & 0xFFFFFFFFull);                  // global_addr[31:0]
  g0.w = (unsigned)((ga >> 32) & 0x1FFFFFFull) | (2u << 30); // addr[56:32]|type=2
  i32x8 g1;
  g1[0] = 0x00020000;                                     // data_size=2 (4B)
  g1[1] = (int)((dim0 & 0xFFFFu) << 16);                  // tensor_dim0[15:0]
  g1[2] = (int)((dim0 >> 16) | ((dim1 & 0xFFFFu) << 16)); // dim0 hi | dim1 lo
  g1[3] = (int)((dim1 >> 16) | ((dim0 & 0xFFFFu) << 16)); // dim1 hi | tile_dim0
  g1[4] = (int)(dim1 & 0xFFFFu);                          // tile_dim1 | tile_dim2=0
  g1[5] = (int)dim0;                                      // dim0_stride[31:0]
  g1[6] = 0;                                              // stride0 hi | stride1 lo
  g1[7] = 0;
  i32x4 gz = (i32x4){0, 0, 0, 0};
#if __clang_major__ >= 23
  i32x8 gz8 = (i32x8){0, 0, 0, 0, 0, 0, 0, 0};
  __builtin_amdgcn_tensor_load_to_lds(g0, g1, gz, gz, gz8, 0);
#else
  __builtin_amdgcn_tensor_load_to_lds(g0, g1, gz, gz, 0);
#endif
#else
  (void)gptr; (void)lds_byte; (void)dim0; (void)dim1;
#endif
}

template <int RES>
__device__ __forceinline__ float trilerp(const float* __restrict__ cb,
                                         float px, float py, float pz) {
  constexpr float S = (float)(RES - 1);
  float x = px * S, y = py * S, z = pz * S;
  float xf = floorf(x), yf = floorf(y), zf = floorf(z);
  float fx = x - xf, fy = y - yf, fz = z - zf;
  int cx = (int)xf, cy = (int)yf, cz = (int)zf;
  cx = cx < 0 ? 0 : (cx > RES - 1 ? RES - 1 : cx);
  cy = cy < 0 ? 0 : (cy > RES - 1 ? RES - 1 : cy);
  cz = cz < 0 ? 0 : (cz > RES - 1 ? RES - 1 : cz);
  int cx1 = (cx + 1 > RES - 1) ? RES - 1 : cx + 1;
  int cy1 = (cy + 1 > RES - 1) ? RES - 1 : cy + 1;
  int cz1 = (cz + 1 > RES - 1) ? RES - 1 : cz + 1;
  int z0 = cz * (RES * RES), z1 = cz1 * (RES * RES);
  int b00 = z0 + cy * RES, b10 = z0 + cy1 * RES;
  int b01 = z1 + cy * RES, b11 = z1 + cy1 * RES;
  float v000 = cb[b00 + cx], v100 = cb[b00 + cx1];
  float v010 = cb[b10 + cx], v110 = cb[b10 + cx1];
  float v001 = cb[b01 + cx], v101 = cb[b01 + cx1];
  float v011 = cb[b11 + cx], v111 = cb[b11 + cx1];
  float c00 = fmaf(fx, v100 - v000, v000);
  float c10 = fmaf(fx, v110 - v010, v010);
  float c01 = fmaf(fx, v101 - v001, v001);
  float c11 = fmaf(fx, v111 - v011, v011);
  float c0 = fmaf(fy, c10 - c00, c00);
  float c1 = fmaf(fy, c11 - c01, c01);
  return fmaf(fz, c1 - c0, c0);
}

__global__ __launch_bounds__(BLOCK_SIZE)
void DenseGrid_88278757802386_kernel(const float* __restrict__ pts,
                                     const float* __restrict__ cb0,
                                     const float* __restrict__ cb1,
                                     const float* __restrict__ cb2,
                                     const float* __restrict__ cb3,
                                     const float* __restrict__ cb4,
                                     float* __restrict__ out, int n) {
  __shared__ float s_cb0[16 * 16 * 16];   //  16 KB
  __shared__ float s_cb1[32 * 32 * 32];   // 128 KB  (total 144 KB <= 320 KB/WGP)

#if HAVE_TDM
  if ((threadIdx.x >> 5) == 0) {          // wave 0 issues the TDM DMAs
    tdm_load_2d_f32(cb0, (unsigned)(uintptr_t)(void*)s_cb0, 4096u, 1u);
    tdm_load_2d_f32(cb1, (unsigned)(uintptr_t)(void*)s_cb1, 8192u, 4u);
#if __has_builtin(__builtin_amdgcn_s_wait_tensorcnt)
    __builtin_amdgcn_s_wait_tensorcnt(0);
#else
    asm volatile("s_wait_tensorcnt 0" ::: "memory");
#endif
  }
#else
  for (int i = threadIdx.x; i < (16 * 16 * 16) / 4; i += BLOCK_SIZE)
    ((float4*)s_cb0)[i] = ((const float4*)cb0)[i];
  for (int i = threadIdx.x; i < (32 * 32 * 32) / 4; i += BLOCK_SIZE)
    ((float4*)s_cb1)[i] = ((const float4*)cb1)[i];
#endif
  __syncthreads();

  int stride = (int)(gridDim.x * blockDim.x);
  for (int i = (int)(blockIdx.x * blockDim.x + threadIdx.x); i < n; i += stride) {
    // Streamed once: non-temporal so the point stream never evicts the
    // L2-resident codebooks (76 MB of 192 MB L2).
    float x = __builtin_nontemporal_load(pts + 3 * i + 0);
    float y = __builtin_nontemporal_load(pts + 3 * i + 1);
    float z = __builtin_nontemporal_load(pts + 3 * i + 2);
    float px = fmaf(x, 0.5f, 0.5f);
    float py = fmaf(y, 0.5f, 0.5f);
    float pz = fmaf(z, 0.5f, 0.5f);

    float acc = trilerp<16>(s_cb0, px, py, pz);   // LDS gathers
    acc += trilerp<32>(s_cb1, px, py, pz);        // LDS gathers
    acc += trilerp<64>(cb2, px, py, pz);          // L2-resident (1 MB)
    acc += trilerp<128>(cb3, px, py, pz);         // L2-resident (8 MB)
    acc += trilerp<256>(cb4, px, py, pz);         // L2-resident (67 MB)

    __builtin_nontemporal_store(acc, out + i);
  }
}

extern "C" void kernel_launch(void* const* d_in, const int* in_sizes, int n_in,
                              void* d_out, int out_size, void* d_ws, size_t ws_size,
                              hipStream_t stream) {
  (void)n_in; (void)out_size; (void)d_ws; (void)ws_size;
  const float* pts = (const float*)d_in[0];
  const float* cb0 = (const float*)d_in[1];
  const float* cb1 = (const float*)d_in[2];
  const float* cb2 = (const float*)d_in[3];
  const float* cb3 = (const float*)d_in[4];
  const float* cb4 = (const float*)d_in[5];
  float* out = (float*)d_out;
  int n = in_sizes[0] / 3;

  int maxb = (n + BLOCK_SIZE - 1) / BLOCK_SIZE;
  int blocks = 1024;                 // bound redundant LDS-staging traffic
  if (blocks > maxb) blocks = maxb;
  if (blocks < 1) blocks = 1;

  hipLaunchKernelGGL(DenseGrid_88278757802386_kernel, dim3(blocks),
                     dim3(BLOCK_SIZE), 0, stream,
                     pts, cb0, cb1, cb2, cb3, cb4, out, n);
}